// IPL_6347961663550
// MI455X (gfx1250) — compile-verified
//
#include <hip/hip_runtime.h>

// Problem geometry (fixed by the reference).
#define NB   8
#define NC   128
#define NH   192
#define NW   192
#define NH2  96
#define NW2  96
#define NPLANE (NB * NC)        // 1024 (b,c) planes
#define CS   (NH2 * NW2)        // 9216: output channel stride
#define SECT (NB * 2 * NC * NH2 * NW2)  // py section offset in d_out

// Native clang vector types (required by __builtin_nontemporal_*).
typedef float v4f __attribute__((ext_vector_type(4)));
typedef float v2f __attribute__((ext_vector_type(2)));

// block(48,4): threadIdx.x = quad index along the row (4 input cols each),
//              threadIdx.y + 4*blockIdx.x = output row h2 (0..95),
//              blockIdx.y  = plane (b*NC + c, 0..1023).
// No integer division anywhere in the kernel.
__global__ __launch_bounds__(192) void IPL_softpool2x2_kernel(
    const float* __restrict__ px,
    const float* __restrict__ py,
    const float* __restrict__ x,
    float* __restrict__ outpx,
    float* __restrict__ outpy)
{
    const int wq    = threadIdx.x;                       // 0..47
    const int h2    = blockIdx.x * 4 + threadIdx.y;      // 0..95
    const int plane = blockIdx.y;                        // 0..1023

    // ---- Streaming loads: 6x b128 with non-temporal hint (single-use data) ----
    // All share one 32-bit offset; row 1 is +NW floats (imm offset 768 B).
    const int inoff = (plane * NH + 2 * h2) * NW + 4 * wq;
    v4f x0 = __builtin_nontemporal_load((const v4f*)(x  + inoff));
    v4f x1 = __builtin_nontemporal_load((const v4f*)(x  + inoff + NW));
    v4f a0 = __builtin_nontemporal_load((const v4f*)(px + inoff));
    v4f a1 = __builtin_nontemporal_load((const v4f*)(px + inoff + NW));
    v4f b0 = __builtin_nontemporal_load((const v4f*)(py + inoff));
    v4f b1 = __builtin_nontemporal_load((const v4f*)(py + inoff + NW));

    const float LOG2E = 1.4426950408889634f;

    // ---- Block A: cols {0,1} of the quad ----
    float mA  = fmaxf(fmaxf(x0.x, x0.y), fmaxf(x1.x, x1.y));
    float eA0 = __builtin_amdgcn_exp2f((x0.x - mA) * LOG2E);
    float eA1 = __builtin_amdgcn_exp2f((x0.y - mA) * LOG2E);
    float eA2 = __builtin_amdgcn_exp2f((x1.x - mA) * LOG2E);
    float eA3 = __builtin_amdgcn_exp2f((x1.y - mA) * LOG2E);
    float invA = __builtin_amdgcn_rcpf(eA0 + eA1 + eA2 + eA3);  // sum >= 1
    float pxA = (eA0 * a0.x + eA1 * a0.y + eA2 * a1.x + eA3 * a1.y) * invA;
    float pyA = (eA0 * b0.x + eA1 * b0.y + eA2 * b1.x + eA3 * b1.y) * invA;

    // ---- Block B: cols {2,3} of the quad ----
    float mB  = fmaxf(fmaxf(x0.z, x0.w), fmaxf(x1.z, x1.w));
    float eB0 = __builtin_amdgcn_exp2f((x0.z - mB) * LOG2E);
    float eB1 = __builtin_amdgcn_exp2f((x0.w - mB) * LOG2E);
    float eB2 = __builtin_amdgcn_exp2f((x1.z - mB) * LOG2E);
    float eB3 = __builtin_amdgcn_exp2f((x1.w - mB) * LOG2E);
    float invB = __builtin_amdgcn_rcpf(eB0 + eB1 + eB2 + eB3);
    float pxB = (eB0 * a0.z + eB1 * a0.w + eB2 * a1.z + eB3 * a1.w) * invB;
    float pyB = (eB0 * b0.z + eB1 * b0.w + eB2 * b1.z + eB3 * b1.w) * invB;

    // ---- Stores: channel-repeat 2x. One VGPR offset; +CS is an imm offset. ----
    // Output channel layout after jnp.repeat(.,2,axis=1): ch (2c) and (2c+1).
    const int oo = ((plane * 2) * NH2 + h2) * NW2 + 2 * wq;

    v2f vpx = {pxA, pxB};
    v2f vpy = {pyA, pyB};
    __builtin_nontemporal_store(vpx, (v2f*)(outpx + oo));
    __builtin_nontemporal_store(vpx, (v2f*)(outpx + oo + CS));
    __builtin_nontemporal_store(vpy, (v2f*)(outpy + oo));
    __builtin_nontemporal_store(vpy, (v2f*)(outpy + oo + CS));
}

extern "C" void kernel_launch(void* const* d_in, const int* in_sizes, int n_in,
                              void* d_out, int out_size, void* d_ws, size_t ws_size,
                              hipStream_t stream) {
    const float* px = (const float*)d_in[0];
    const float* py = (const float*)d_in[1];
    const float* x  = (const float*)d_in[2];
    float* outpx = (float*)d_out;
    float* outpy = (float*)d_out + SECT;

    (void)in_sizes; (void)n_in; (void)out_size; (void)d_ws; (void)ws_size;

    dim3 block(48, 4);     // 192 threads = 6 wave32
    dim3 grid(NH2 / 4, NPLANE);   // 24 x 1024 workgroups
    IPL_softpool2x2_kernel<<<grid, block, 0, stream>>>(px, py, x, outpx, outpy);
}